// PosClassifier_83253646066046
// MI455X (gfx1250) — compile-verified
//
#include <hip/hip_runtime.h>
#include <hip/hip_bf16.h>
#include <math.h>

#define BB 128
#define NN 512
#define DD 16
#define MM 64
#define KK 6
#define WPB 2            // waves per block in egnn_kernel
#define TPB (WPB * 32)

typedef __attribute__((ext_vector_type(2))) float v2f;
typedef __attribute__((ext_vector_type(8))) float v8f;

static __device__ __forceinline__ v8f wmma4(v2f a, v2f b, v8f c) {
  // V_WMMA_F32_16X16X4_F32: (neg_a, A, neg_b, B, c_mod, C, reuse_a, reuse_b)
  return __builtin_amdgcn_wmma_f32_16x16x4_f32(false, a, false, b, (short)0, c,
                                               false, false);
}

static __device__ __forceinline__ float silu_f(float x) {
  return x / (1.0f + expf(-x));
}

// ---------------------------------------------------------------------------
// Kernel 1: per-batch brute-force KNN (K=6 incl. self), emit rel_xyz + d^2
// ---------------------------------------------------------------------------
__global__ __launch_bounds__(NN) void knn_kernel(const float* __restrict__ pos,
                                                 float* __restrict__ knn) {
  __shared__ float px[NN], py[NN], pz[NN];
  const int b = blockIdx.x;
  const int i = threadIdx.x;
  const float* p = pos + ((size_t)b * NN + i) * 3;
  const float x = p[0], y = p[1], z = p[2];
  px[i] = x; py[i] = y; pz[i] = z;
  __syncthreads();

  float bd[KK];
  int   bj[KK];
#pragma unroll
  for (int k = 0; k < KK; ++k) { bd[k] = 3.0e38f; bj[k] = 0; }

  for (int j = 0; j < NN; ++j) {
    const float dx = x - px[j];
    const float dy = y - py[j];
    const float dz = z - pz[j];
    const float d = dx * dx + dy * dy + dz * dz;
    if (d < bd[KK - 1]) {            // strict <  -> earlier index wins ties
      bd[KK - 1] = d; bj[KK - 1] = j;
#pragma unroll
      for (int q = KK - 1; q > 0; --q) {
        if (bd[q] < bd[q - 1]) {
          float td = bd[q]; bd[q] = bd[q - 1]; bd[q - 1] = td;
          int   tj = bj[q]; bj[q] = bj[q - 1]; bj[q - 1] = tj;
        }
      }
    }
  }

  float* o = knn + (((size_t)b * NN + i) * KK) * 4;
#pragma unroll
  for (int k = 0; k < KK; ++k) {
    const int j = bj[k];
    o[k * 4 + 0] = x - px[j];
    o[k * 4 + 1] = y - py[j];
    o[k * 4 + 2] = z - pz[j];
    o[k * 4 + 3] = bd[k];
  }
}

// ---------------------------------------------------------------------------
// Kernel 2: edge MLP (WMMA fp32) + gate + message sum + node MLP + pool-atomics
// One wave handles 8 nodes = 48 edges = 3 tiles of 16 edges.
// ---------------------------------------------------------------------------
struct WaveLds {
  float ein[16][8];    // 16 edges x padded 8 features
  float h[16][80];     // layer1 activations (74 -> padded 80)
  float m[16][64];     // layer2 activations (messages)
  float gate[16];
  float mi[8][64];     // per-node message sum
  float hn[8][32];     // node-MLP hidden
  float fsum[32];      // pooled partials
};

__global__ __launch_bounds__(TPB) void egnn_kernel(
    const float* __restrict__ knn,
    const float* __restrict__ e1w, const float* __restrict__ e1b,
    const float* __restrict__ e2w, const float* __restrict__ e2b,
    const float* __restrict__ gwp, const float* __restrict__ gbp,
    const float* __restrict__ n1w, const float* __restrict__ n1b,
    const float* __restrict__ n2w, const float* __restrict__ n2b,
    float* __restrict__ pooled) {
  __shared__ float W1[8][80];      // e1.w rows 32..36, zero padded
  __shared__ float B1v[80];
  __shared__ float W2[80][64];     // e2.w, zero padded rows 74..79
  __shared__ float B2v[64];
  __shared__ float GW[64];
  __shared__ float N1W[64][32];    // n1.w rows 16..79 (feats part is zero)
  __shared__ float N1Bv[32];
  __shared__ float N2W[32][16];
  __shared__ float N2Bv[16];
  __shared__ WaveLds wl[WPB];

  const int tid = threadIdx.x;

  for (int idx = tid; idx < 8 * 80; idx += TPB) {
    int r = idx / 80, c = idx % 80;
    W1[r][c] = (r < 5 && c < 74) ? e1w[(32 + r) * 74 + c] : 0.0f;
  }
  for (int idx = tid; idx < 80; idx += TPB) B1v[idx] = (idx < 74) ? e1b[idx] : 0.0f;
  for (int idx = tid; idx < 80 * 64; idx += TPB) {
    int r = idx / 64, c = idx % 64;
    W2[r][c] = (r < 74) ? e2w[r * 64 + c] : 0.0f;
  }
  for (int idx = tid; idx < 64; idx += TPB) { B2v[idx] = e2b[idx]; GW[idx] = gwp[idx]; }
  for (int idx = tid; idx < 64 * 32; idx += TPB) {
    int r = idx / 32, c = idx % 32;
    N1W[r][c] = n1w[(16 + r) * 32 + c];
  }
  for (int idx = tid; idx < 32; idx += TPB) N1Bv[idx] = n1b[idx];
  for (int idx = tid; idx < 32 * 16; idx += TPB) N2W[idx / 16][idx % 16] = n2w[idx];
  for (int idx = tid; idx < 16; idx += TPB) N2Bv[idx] = n2b[idx];

  const int wid = tid >> 5;
  const int lane = tid & 31;
  WaveLds& L = wl[wid];
  {
    float* mip = &L.mi[0][0];
    for (int idx = lane; idx < 8 * 64; idx += 32) mip[idx] = 0.0f;
  }
  __syncthreads();

  const int gwave = blockIdx.x * WPB + wid;
  const int node0 = gwave * 8;          // first global node for this wave
  const int batch = node0 >> 9;         // node0 / 512 (8 | 512 -> uniform)
  const float gb = gbp[0];

  const int half = lane >> 4;           // 0: M/K-lo half, 1: hi half
  const int l15 = lane & 15;
  const int r0 = half * 8;              // D rows owned by this lane
  const v8f zero8 = {0.f, 0.f, 0.f, 0.f, 0.f, 0.f, 0.f, 0.f};

  for (int st = 0; st < 3; ++st) {
    // ---- build edge_in tile (fourier distance features) ----
    if (lane < 16) {
      const size_t eg = (size_t)node0 * KK + (size_t)st * 16 + lane;
      const float d = knn[eg * 4 + 3];
      L.ein[lane][0] = sinf(d);
      L.ein[lane][1] = sinf(d * 0.5f);
      L.ein[lane][2] = cosf(d);
      L.ein[lane][3] = cosf(d * 0.5f);
      L.ein[lane][4] = d;
      L.ein[lane][5] = 0.0f;
      L.ein[lane][6] = 0.0f;
      L.ein[lane][7] = 0.0f;
    }
    __syncthreads();

    // ---- layer1: [16x8] @ [8x80] via WMMA f32 16x16x4 ----
    v8f acc1[5];
#pragma unroll
    for (int t = 0; t < 5; ++t) acc1[t] = zero8;
#pragma unroll
    for (int s = 0; s < 2; ++s) {
      const int kb = s * 4 + half * 2;
      v2f a = *(const v2f*)&L.ein[l15][kb];
#pragma unroll
      for (int t = 0; t < 5; ++t) {
        v2f bf;
        bf.x = W1[kb][t * 16 + l15];
        bf.y = W1[kb + 1][t * 16 + l15];
        acc1[t] = wmma4(a, bf, acc1[t]);
      }
    }
#pragma unroll
    for (int t = 0; t < 5; ++t) {
      const int n = t * 16 + l15;
      const float bias = B1v[n];
#pragma unroll
      for (int v = 0; v < 8; ++v) L.h[r0 + v][n] = silu_f(acc1[t][v] + bias);
    }
    __syncthreads();

    // ---- layer2: [16x80] @ [80x64] via WMMA ----
    v8f acc2[4];
#pragma unroll
    for (int t = 0; t < 4; ++t) acc2[t] = zero8;
    for (int s = 0; s < 20; ++s) {
      const int kb = s * 4 + half * 2;
      v2f a = *(const v2f*)&L.h[l15][kb];
#pragma unroll
      for (int t = 0; t < 4; ++t) {
        v2f bf;
        bf.x = W2[kb][t * 16 + l15];
        bf.y = W2[kb + 1][t * 16 + l15];
        acc2[t] = wmma4(a, bf, acc2[t]);
      }
    }
#pragma unroll
    for (int t = 0; t < 4; ++t) {
      const int n = t * 16 + l15;
      const float bias = B2v[n];
#pragma unroll
      for (int v = 0; v < 8; ++v) L.m[r0 + v][n] = silu_f(acc2[t][v] + bias);
    }
    __syncthreads();

    // ---- soft edge gate: sigmoid(m . gw + gb) ----
    if (lane < 16) {
      float acc = gb;
      for (int c = 0; c < 64; ++c) acc += L.m[lane][c] * GW[c];
      L.gate[lane] = 1.0f / (1.0f + expf(-acc));
    }
    __syncthreads();

    // ---- gated accumulation into per-node mi (lane owns cols lane, lane+32) ----
    {
      const int c0 = lane, c1 = lane + 32;
#pragma unroll
      for (int e = 0; e < 16; ++e) {
        const int n = (st * 16 + e) / KK;     // 0..7, uniform per iteration
        const float g = L.gate[e];
        L.mi[n][c0] += g * L.m[e][c0];
        L.mi[n][c1] += g * L.m[e][c1];
      }
    }
    __syncthreads();
  }

  // ---- node MLP layer1: mi[8][64] @ [64][32], output col = lane ----
#pragma unroll
  for (int n = 0; n < 8; ++n) {
    float acc = N1Bv[lane];
    for (int k = 0; k < 64; ++k) acc += L.mi[n][k] * N1W[k][lane];
    L.hn[n][lane] = silu_f(acc);
  }
  __syncthreads();

  // ---- node MLP layer2 + per-wave pooled partials ----
  {
    const int o = lane & 15;
    float ps = 0.0f;
    for (int nn = half; nn < 8; nn += 2) {
      float acc = N2Bv[o];
#pragma unroll
      for (int k = 0; k < 32; ++k) acc += L.hn[nn][k] * N2W[k][o];
      ps += acc;                    // residual +feats is zero
    }
    L.fsum[lane] = ps;
  }
  __syncthreads();
  if (lane < 16) {
    atomicAdd(&pooled[batch * 16 + lane], L.fsum[lane] + L.fsum[lane + 16]);
  }
}

// ---------------------------------------------------------------------------
// Kernel 3: mean pool + classifier head 16 -> 64 -> 128 -> 1
// ---------------------------------------------------------------------------
__global__ __launch_bounds__(64) void head_kernel(
    const float* __restrict__ pooled,
    const float* __restrict__ h1w, const float* __restrict__ h1b,
    const float* __restrict__ h2w, const float* __restrict__ h2b,
    const float* __restrict__ h3w, const float* __restrict__ h3b,
    float* __restrict__ out) {
  const int b = blockIdx.x * blockDim.x + threadIdx.x;
  if (b >= BB) return;
  float p[16];
#pragma unroll
  for (int d = 0; d < 16; ++d) p[d] = pooled[b * 16 + d] * (1.0f / (float)NN);
  float x1[64];
  for (int o = 0; o < 64; ++o) {
    float acc = h1b[o];
#pragma unroll
    for (int k = 0; k < 16; ++k) acc += p[k] * h1w[k * 64 + o];
    x1[o] = fmaxf(acc, 0.0f);
  }
  float res = h3b[0];
  for (int o = 0; o < 128; ++o) {
    float acc = h2b[o];
    for (int k = 0; k < 64; ++k) acc += x1[k] * h2w[k * 128 + o];
    res += fmaxf(acc, 0.0f) * h3w[o];
  }
  out[b] = res;
}

// ---------------------------------------------------------------------------
extern "C" void kernel_launch(void* const* d_in, const int* in_sizes, int n_in,
                              void* d_out, int out_size, void* d_ws, size_t ws_size,
                              hipStream_t stream) {
  (void)in_sizes; (void)n_in; (void)out_size; (void)ws_size;
  const float* pos = (const float*)d_in[0];
  const float* e1w = (const float*)d_in[1];
  const float* e1b = (const float*)d_in[2];
  const float* e2w = (const float*)d_in[3];
  const float* e2b = (const float*)d_in[4];
  const float* gw  = (const float*)d_in[5];
  const float* gb  = (const float*)d_in[6];
  // d_in[7] coors_scale, d_in[8..11] c1/c2: dead code (coors_out unused)
  const float* n1w = (const float*)d_in[12];
  const float* n1b = (const float*)d_in[13];
  const float* n2w = (const float*)d_in[14];
  const float* n2b = (const float*)d_in[15];
  const float* h1w = (const float*)d_in[16];
  const float* h1b = (const float*)d_in[17];
  const float* h2w = (const float*)d_in[18];
  const float* h2b = (const float*)d_in[19];
  const float* h3w = (const float*)d_in[20];
  const float* h3b = (const float*)d_in[21];

  float* knn = (float*)d_ws;
  const size_t knn_bytes = (size_t)BB * NN * KK * 4 * sizeof(float);
  float* pooled = (float*)((char*)d_ws + knn_bytes);

  hipMemsetAsync(pooled, 0, (size_t)BB * 16 * sizeof(float), stream);
  knn_kernel<<<BB, NN, 0, stream>>>(pos, knn);
  egnn_kernel<<<(BB * NN) / (8 * WPB), TPB, 0, stream>>>(
      knn, e1w, e1b, e2w, e2b, gw, gb, n1w, n1b, n2w, n2b, pooled);
  head_kernel<<<2, 64, 0, stream>>>(pooled, h1w, h1b, h2w, h2b, h3w, h3b,
                                    (float*)d_out);
}